// EncoderMemNN_2010044695259
// MI455X (gfx1250) — compile-verified
//
#include <hip/hip_runtime.h>

#define M_SLOTS 2048
#define BATCH   64
#define TOKENS  4
#define DIM     128
#define VOCAB   50000
#define HOPS    3
#define OUT_CHUNKS 16   // blocks per batch row in out_kernel; each covers M/16 = 128 slots

typedef __attribute__((ext_vector_type(2))) float v2f;
typedef __attribute__((ext_vector_type(4))) float v4f;
typedef __attribute__((ext_vector_type(8))) float v8f;

__global__ __launch_bounds__(256) void zero_kernel(float* __restrict__ p, int n) {
  int i = blockIdx.x * blockDim.x + threadIdx.x;
  if (i < n) p[i] = 0.0f;
}

// scores[b][tile..tile+15] = (sum_t mask * C[tok][:]) . u[b][:]
// One wave32 per 16-slot tile; K-reduction via chained V_WMMA_F32_16X16X4_F32.
__global__ __launch_bounds__(32) void score_kernel(
    const float* __restrict__ Ct,     // embedding table for this hop [VOCAB][DIM]
    const int*   __restrict__ story,  // [M][B][T]
    const float* __restrict__ u,      // [B][DIM]
    float*       __restrict__ scores) // [B][M]
{
  const int lane = threadIdx.x;       // 0..31
  const int half = lane >> 4;         // 0: K0,K1  1: K2,K3  (A-matrix 16x4 f32 layout)
  const int row  = lane & 15;         // slot within tile (M index of A)
  const int tile = blockIdx.x * 16;
  const int b    = blockIdx.y;
  const int slot = tile + row;

  // Tokens of this lane's slot (lanes l and l+16 duplicate the 4 small loads; rows they
  // gather are disjoint dim-halves, so embedding-row bytes are read exactly once).
  int   tok[TOKENS];
  float w[TOKENS];
  const int sbase = (slot * BATCH + b) * TOKENS;
#pragma unroll
  for (int t = 0; t < TOKENS; ++t) {
    tok[t] = story[sbase + t];
    w[t]   = (tok[t] != 0) ? 1.0f : 0.0f;  // padding_idx=0: multiply, keep EXEC uniform for WMMA
  }

  const float* ub = u + b * DIM;
  v8f acc = {};
#pragma unroll 4
  for (int c = 0; c < DIM / 4; ++c) {
    const int d0 = c * 4 + half * 2;   // this lane's two K dims of the 4-wide chunk
    float a0 = 0.0f, a1 = 0.0f;
#pragma unroll
    for (int t = 0; t < TOKENS; ++t) {
      const v2f e = *(const v2f*)(Ct + (size_t)tok[t] * DIM + d0);
      a0 += w[t] * e.x;
      a1 += w[t] * e.y;
    }
    v2f A;  A.x = a0; A.y = a1;                 // A[row][K] tile of keys
    v2f Bm = *(const v2f*)(ub + d0);            // B[K][n] = u[d] replicated over n
    acc = __builtin_amdgcn_wmma_f32_16x16x4_f32(
        false, A, false, Bm, (short)0, acc, false, false);
  }

  // D[i][n] identical for all n. C/D layout: lane 0 holds M=0..7 (VGPR 0..7) at N=0,
  // lane 16 holds M=8..15 at N=0.
  float* srow = scores + (size_t)b * M_SLOTS + tile;
  if (lane == 0) {
#pragma unroll
    for (int r = 0; r < 8; ++r) srow[r] = acc[r];
  } else if (lane == 16) {
#pragma unroll
    for (int r = 0; r < 8; ++r) srow[8 + r] = acc[r];
  }
}

// In-place softmax over M=2048 per batch row. One 256-thread block per row.
__global__ __launch_bounds__(256) void softmax_kernel(float* __restrict__ scores) {
  const int b   = blockIdx.x;
  const int tid = threadIdx.x;
  float* s = scores + (size_t)b * M_SLOTS;
  __shared__ float red[256];

  float local[M_SLOTS / 256];
  float mx = -__builtin_inff();
#pragma unroll
  for (int i = 0; i < M_SLOTS / 256; ++i) {
    local[i] = s[tid + i * 256];
    mx = fmaxf(mx, local[i]);
  }
  red[tid] = mx;
  __syncthreads();
  for (int off = 128; off > 0; off >>= 1) {
    if (tid < off) red[tid] = fmaxf(red[tid], red[tid + off]);
    __syncthreads();
  }
  mx = red[0];
  __syncthreads();

  float sum = 0.0f;
#pragma unroll
  for (int i = 0; i < M_SLOTS / 256; ++i) {
    local[i] = __expf(local[i] - mx);
    sum += local[i];
  }
  red[tid] = sum;
  __syncthreads();
  for (int off = 128; off > 0; off >>= 1) {
    if (tid < off) red[tid] += red[tid + off];
    __syncthreads();
  }
  const float inv = 1.0f / red[0];
#pragma unroll
  for (int i = 0; i < M_SLOTS / 256; ++i) s[tid + i * 256] = local[i] * inv;
}

// u[b][:] += sum_m prob[b][m] * (sum_t mask * C[tok][:])
// Lane owns a float4 dim-chunk; slot index is wave-uniform so token/prob loads are scalar.
__global__ __launch_bounds__(256) void out_kernel(
    const float* __restrict__ Ct,     // table for hop+1
    const int*   __restrict__ story,
    const float* __restrict__ prob,   // [B][M]
    float*       __restrict__ u)      // [B][DIM], accumulated
{
  const int b    = blockIdx.y;
  const int wave = threadIdx.x >> 5;
  const int lane = threadIdx.x & 31;
  const int d0   = lane * 4;
  const int slot0 = blockIdx.x * (M_SLOTS / OUT_CHUNKS) + wave * (M_SLOTS / OUT_CHUNKS / 8);

  v4f acc = {};
  for (int s = 0; s < M_SLOTS / OUT_CHUNKS / 8; ++s) {
    const int m = slot0 + s;
    const float p = prob[(size_t)b * M_SLOTS + m];
    const int sbase = (m * BATCH + b) * TOKENS;
#pragma unroll
    for (int t = 0; t < TOKENS; ++t) {
      const int tok = story[sbase + t];
      if (tok != 0) {  // wave-uniform scalar branch, skips dead gathers
        const v4f e = *(const v4f*)(Ct + (size_t)tok * DIM + d0);
        acc += p * e;
      }
    }
  }

  __shared__ float red[8 * DIM];
  *(v4f*)(red + wave * DIM + d0) = acc;
  __syncthreads();
  if (threadIdx.x < DIM) {
    float sum = 0.0f;
#pragma unroll
    for (int wv = 0; wv < 8; ++wv) sum += red[wv * DIM + threadIdx.x];
    atomicAdd(u + (size_t)b * DIM + threadIdx.x, sum);
  }
}

extern "C" void kernel_launch(void* const* d_in, const int* in_sizes, int n_in,
                              void* d_out, int out_size, void* d_ws, size_t ws_size,
                              hipStream_t stream) {
  (void)in_sizes; (void)n_in; (void)out_size; (void)ws_size;
  const int*   story = (const int*)d_in[0];     // [2048][64][4]
  const float* C     = (const float*)d_in[1];   // [4][50000][128]
  float* u      = (float*)d_out;                // [64][128] accumulator == output
  float* scores = (float*)d_ws;                 // [64][2048] = 512 KB scratch

  zero_kernel<<<(BATCH * DIM + 255) / 256, 256, 0, stream>>>(u, BATCH * DIM);

  const size_t tsz = (size_t)VOCAB * DIM;
  for (int hop = 0; hop < HOPS; ++hop) {
    score_kernel<<<dim3(M_SLOTS / 16, BATCH), 32, 0, stream>>>(
        C + (size_t)hop * tsz, story, u, scores);
    softmax_kernel<<<BATCH, 256, 0, stream>>>(scores);
    out_kernel<<<dim3(OUT_CHUNKS, BATCH), 256, 0, stream>>>(
        C + (size_t)(hop + 1) * tsz, story, scores, u);
  }
}